// LearnablePositionalEncoding_63694365000266
// MI455X (gfx1250) — compile-verified
//
#include <hip/hip_runtime.h>
#include <hip/hip_bf16.h>

typedef __attribute__((ext_vector_type(2))) float v2f;
typedef __attribute__((ext_vector_type(8))) float v8f;

#define B_ 8
#define D_ 256
#define N_ 2048
#define WAVES 4   // waves per workgroup in the argmin kernel
#define JT 4      // j-tiles processed concurrently per wave (A-fragment reuse)

// ---------------------------------------------------------------------------
// Kernel 1: sq[b*N + n] = sum_d x[b,d,n]^2   (x layout: (B, D, N))
// ---------------------------------------------------------------------------
__global__ void sq_norm_kernel(const float* __restrict__ x, float* __restrict__ sq) {
  int idx = blockIdx.x * blockDim.x + threadIdx.x;   // b*N + n
  if (idx >= B_ * N_) return;
  int b = idx / N_;
  int n = idx - b * N_;
  const float* xp = x + (size_t)b * D_ * N_ + n;
  float s = 0.0f;
#pragma unroll 8
  for (int d = 0; d < D_; ++d) {
    float v = xp[(size_t)d * N_];
    s += v * v;
  }
  sq[idx] = s;
}

// ---------------------------------------------------------------------------
// Kernel 2: fused Gram (fp32 WMMA) + per-row argmin -> angle[b*N + i]
//
// Each wave owns a 16-row i-block of batch b and sweeps j in blocks of
// 16*JT columns. Per k-step one A fragment feeds JT WMMAs into JT
// independent accumulators (2.5 loads per WMMA instead of 4, and no
// accumulator RAW chaining between consecutive WMMAs).
//
// V_WMMA_F32_16X16X4_F32 keeps the gram in full fp32 (it feeds an argmin).
//
// Fragment layout (ISA 7.12.2, 32-bit A 16x4): lane l holds
//   VGPR0 = A[l%16, 2*(l/16)]   VGPR1 = A[l%16, 2*(l/16)+1]
// B (4x16) mirrors it transposed, so both fragments load identically from
// 16 consecutive floats per half-wave in the (B,D,N) layout -> coalesced.
//
// C layout: VGPR r holds C[m, n] with m = r + 8*(lane/16), n = lane%16.
// Score(i,j) = sq[j] - 2*C  (monotonic equivalent of the reference distance).
// ---------------------------------------------------------------------------
__global__ __launch_bounds__(WAVES * 32)
void argmin_kernel(const float* __restrict__ x,
                   const float* __restrict__ sq,
                   int* __restrict__ angle) {
  const int lane = threadIdx.x & 31;
  const int wave = threadIdx.x >> 5;
  const int b    = blockIdx.y;
  const int i0   = (blockIdx.x * WAVES + wave) * 16;
  const int half = lane >> 4;   // 0: lanes 0-15, 1: lanes 16-31
  const int l16  = lane & 15;

  const float* X   = x  + (size_t)b * D_ * N_;   // X[d*N + n] = xt[n, d]
  const float* sqb = sq + (size_t)b * N_;

  // A-fragment base: rows i0..i0+15, k offset 2*half
  const float* Ap = X + (size_t)(2 * half) * N_ + i0 + l16;

  float bestS[8];
  int   bestJ[8];
#pragma unroll
  for (int r = 0; r < 8; ++r) { bestS[r] = __builtin_inff(); bestJ[r] = 0; }

  for (int j0 = 0; j0 < N_; j0 += 16 * JT) {
    const float* Bp = X + (size_t)(2 * half) * N_ + j0 + l16;

    v8f c[JT];
#pragma unroll
    for (int t = 0; t < JT; ++t) c[t] = (v8f){};

#pragma unroll 2
    for (int k0 = 0; k0 < D_; k0 += 4) {
      v2f a;
      a.x = Ap[(size_t)k0 * N_];
      a.y = Ap[(size_t)k0 * N_ + N_];
#pragma unroll
      for (int t = 0; t < JT; ++t) {
        v2f bf;
        bf.x = Bp[(size_t)k0 * N_ + 16 * t];
        bf.y = Bp[(size_t)k0 * N_ + N_ + 16 * t];
        // D = A(16x4,f32) * B(4x16,f32) + C(16x16,f32)
        c[t] = __builtin_amdgcn_wmma_f32_16x16x4_f32(
                   /*neg_a=*/false, a, /*neg_b=*/false, bf,
                   /*c_mod=*/(short)0, c[t], /*reuse_a=*/false, /*reuse_b=*/false);
      }
    }

    // Consume the JT tiles in increasing-j order (first-index tie-break).
#pragma unroll
    for (int t = 0; t < JT; ++t) {
      const float sj = sqb[j0 + 16 * t + l16];
      const int   j  = j0 + 16 * t + l16;
#pragma unroll
      for (int r = 0; r < 8; ++r) {
        const int i = i0 + r + 8 * half;
        float s = sj - 2.0f * c[t][r];
        if (j == i) s = __builtin_inff();        // mask self-match (diagonal)
        bool better = (s < bestS[r]) || (s == bestS[r] && j < bestJ[r]);
        if (better) { bestS[r] = s; bestJ[r] = j; }
      }
    }
  }

  // Reduce over the 16 columns held by each 16-lane half (xor masks 1..8
  // never cross the half boundary in wave32).
#pragma unroll
  for (int mask = 1; mask <= 8; mask <<= 1) {
#pragma unroll
    for (int r = 0; r < 8; ++r) {
      float os = __shfl_xor(bestS[r], mask, 32);
      int   oj = __shfl_xor(bestJ[r], mask, 32);
      bool better = (os < bestS[r]) || (os == bestS[r] && oj < bestJ[r]);
      if (better) { bestS[r] = os; bestJ[r] = oj; }
    }
  }

  if (l16 == 0) {
#pragma unroll
    for (int r = 0; r < 8; ++r) {
      const int i = i0 + r + 8 * half;   // half=0 writes rows 0-7, half=1 rows 8-15
      int d = i - bestJ[r];
      if (d < 0) d = -d;
      angle[b * N_ + i] = d;             // angle in [0, N)
    }
  }
}

// ---------------------------------------------------------------------------
// Kernel 3: out[b,d,n] = x[b,d,n] + emb[angle[b,n], d]
// ---------------------------------------------------------------------------
__global__ void add_emb_kernel(const float* __restrict__ x,
                               const float* __restrict__ emb,
                               const int* __restrict__ angle,
                               float* __restrict__ out) {
  size_t idx = (size_t)blockIdx.x * blockDim.x + threadIdx.x;
  if (idx >= (size_t)B_ * D_ * N_) return;
  int n = (int)(idx % N_);
  size_t bd = idx / N_;
  int d = (int)(bd % D_);
  int b = (int)(bd / D_);
  int a = angle[b * N_ + n];
  out[idx] = x[idx] + emb[(size_t)a * D_ + d];
}

// ---------------------------------------------------------------------------
extern "C" void kernel_launch(void* const* d_in, const int* in_sizes, int n_in,
                              void* d_out, int out_size, void* d_ws, size_t ws_size,
                              hipStream_t stream) {
  (void)in_sizes; (void)n_in; (void)out_size; (void)ws_size;

  const float* x   = (const float*)d_in[0];   // (B, D, N) fp32
  const float* emb = (const float*)d_in[1];   // (N, D)    fp32
  float*       out = (float*)d_out;           // (B, D, N) fp32

  float* sq    = (float*)d_ws;                                   // B*N floats
  int*   angle = (int*)((char*)d_ws + (size_t)B_ * N_ * sizeof(float)); // B*N ints

  // 1) squared norms
  sq_norm_kernel<<<(B_ * N_ + 255) / 256, 256, 0, stream>>>(x, sq);

  // 2) fused fp32-WMMA gram + argmin -> angles
  dim3 grid2(N_ / (16 * WAVES), B_);
  argmin_kernel<<<grid2, WAVES * 32, 0, stream>>>(x, sq, angle);

  // 3) gather embedding row and add
  const int total = B_ * D_ * N_;
  add_emb_kernel<<<(total + 255) / 256, 256, 0, stream>>>(x, emb, angle, out);
}